// IF_33122787787165
// MI455X (gfx1250) — compile-verified
//
#include <hip/hip_runtime.h>

// Integrate-and-Fire forward: x (256,64,56,56) f32 viewed as (B=32, T=8, N=200704).
// Pure streaming workload: 411 MB total @ 23.3 TB/s => ~17.6 us floor. No matmul
// structure => WMMA not applicable; optimize the data-movement path instead:
// wave32 b128 accesses, async global->LDS staging (ASYNCcnt), NT stores.

#define T_STEPS 8
#define VTH_F 1.0f
#define N_INNER 200704                 // C*H*W = 64*56*56
#define BLOCK 256
#define VEC 4
#define CHUNK (BLOCK * VEC)            // 1024 floats per timestep per block
#define TILES_PER_B (N_INNER / CHUNK)  // 196

typedef __attribute__((ext_vector_type(4))) float v4f;
// Matches the builtin's parameter type: vector_size(16) int pointer.
typedef int v4i __attribute__((vector_size(4 * sizeof(int))));

#if defined(__HIP_DEVICE_COMPILE__) && \
    __has_builtin(__builtin_amdgcn_global_load_async_to_lds_b128)
#define IF_ASYNC_LDS 1
#else
#define IF_ASYNC_LDS 0
#endif

__global__ __launch_bounds__(BLOCK) void if_spike_kernel(
    const float* __restrict__ x, float* __restrict__ out) {
  const int tid = threadIdx.x;
  const int tile = blockIdx.x;
  const int b = tile / TILES_PER_B;
  const int chunk = tile - b * TILES_PER_B;
  const int off = chunk * CHUNK + tid * VEC;  // float offset inside one (b,t) image

  v4f xv[T_STEPS];

#if IF_ASYNC_LDS
  // Stage all 8 timestep slices through LDS with async copies. Each lane writes
  // and reads only its own 16B/timestep region -> per-wave ASYNCcnt ordering is
  // sufficient, no workgroup barrier required. 8 transfers in flight per wave.
  __shared__ v4f lbuf[T_STEPS * BLOCK];  // 32 KB/block (320 KB/WGP -> 10 blocks)
#pragma unroll
  for (int t = 0; t < T_STEPS; ++t) {
    const float* g = x + (size_t)(b * T_STEPS + t) * N_INNER + off;
    __builtin_amdgcn_global_load_async_to_lds_b128(
        (v4i*)g, (v4i*)&lbuf[t * BLOCK + tid], /*offset=*/0, /*cpol=*/0);
  }
  __asm__ __volatile__("" ::: "memory");
#if __has_builtin(__builtin_amdgcn_s_wait_asynccnt)
  __builtin_amdgcn_s_wait_asynccnt(0);
#else
  __asm__ __volatile__("s_wait_asynccnt 0x0" ::: "memory");
#endif
  __asm__ __volatile__("" ::: "memory");
#pragma unroll
  for (int t = 0; t < T_STEPS; ++t) {
    xv[t] = lbuf[t * BLOCK + tid];  // ds_load_b128
  }
#else
  // Fallback: direct b128 non-temporal loads (no reuse; input stream > L2).
#pragma unroll
  for (int t = 0; t < T_STEPS; ++t) {
    xv[t] = __builtin_nontemporal_load(
        (const v4f*)(x + (size_t)(b * T_STEPS + t) * N_INNER + off));
  }
#endif

  // Sequential IF recurrence over T=8, fully register-resident.
  v4f mem = {0.0f, 0.0f, 0.0f, 0.0f};
#pragma unroll
  for (int t = 0; t < T_STEPS; ++t) {
    v4f o;
#pragma unroll
    for (int i = 0; i < 4; ++i) {
      float m = mem[i] + xv[t][i];          // mem*DECAY + x, DECAY==1
      float s = (m > VTH_F) ? 1.0f : 0.0f;  // Heaviside(mem - VTH)
      o[i] = s;
      mem[i] = (s != 0.0f) ? 0.0f : m;      // hard reset
    }
    // Write-once output, never re-read -> non-temporal b128 store.
    __builtin_nontemporal_store(
        o, (v4f*)(out + (size_t)(b * T_STEPS + t) * N_INNER + off));
  }
}

extern "C" void kernel_launch(void* const* d_in, const int* in_sizes, int n_in,
                              void* d_out, int out_size, void* d_ws, size_t ws_size,
                              hipStream_t stream) {
  (void)n_in; (void)d_ws; (void)ws_size; (void)in_sizes;
  const float* x = (const float*)d_in[0];
  float* out = (float*)d_out;
  // out_size = 32*8*200704 = 51,380,224 floats; 8192 floats handled per block.
  const int tiles = out_size / (T_STEPS * CHUNK);  // 6272
  if_spike_kernel<<<tiles, BLOCK, 0, stream>>>(x, out);
}